// Conv_RBS_density_I2_3D_1632087572681
// MI455X (gfx1250) — compile-verified
//
#include <hip/hip_runtime.h>

typedef __attribute__((ext_vector_type(2))) float v2f;
typedef __attribute__((ext_vector_type(8))) float v8f;

#define DDIM 1024
#define NGATES 18

// ---------------------------------------------------------------------------
// Kernel 1: build W = U17 * U16 * ... * U0.
// Left-multiplying by a gate only mixes ROWS of W, independently per column,
// so each block owns a 1024 x 64 column stripe held entirely in 256KB LDS
// (CDNA5: up to 320KB per workgroup). 16 blocks cover all 1024 columns.
// ---------------------------------------------------------------------------
__global__ __launch_bounds__(256) void build_w_kernel(
    const float* __restrict__ thetas, float* __restrict__ W) {
  extern __shared__ float S[];  // [1024][64] stripe, row-major, stride 64
  const int tid = threadIdx.x;
  const int c0  = blockIdx.x * 64;

  // stripe = columns [c0, c0+64) of the identity
  for (int idx = tid; idx < 1024 * 64; idx += 256) {
    const int row = idx >> 6, cl = idx & 63;
    S[idx] = (row == c0 + cl) ? 1.0f : 0.0f;
  }
  __syncthreads();

  const int pa[6] = {0, 0, 0, 1, 1, 2};
  const int pb[6] = {1, 2, 3, 2, 3, 3};

  for (int g = 0; g < NGATES; ++g) {
    const int axis = g / 6;            // 0=row, 1=col, 2=ch (gate list order)
    const int a = pa[g % 6], b = pb[g % 6];
    const float th  = thetas[g];
    const float cth = __cosf(th);
    const float sth = __sinf(th);
    const int P = (axis == 2) ? 256 : 64;   // disjoint (ia,ib) row pairs
    for (int idx = tid; idx < P * 64; idx += 256) {
      const int p = idx >> 6, cl = idx & 63;
      int ia, ib;
      if (axis == 0) {                      // flat = i*64 + j*4 + ch, i in {a,b}
        ia = a * 64 + p;            ib = b * 64 + p;
      } else if (axis == 1) {               // j in {a,b}
        const int r = p >> 2, ch = p & 3;
        ia = r * 64 + a * 4 + ch;   ib = r * 64 + b * 4 + ch;
      } else {                              // ch in {a,b}
        const int r = p >> 4, cc = p & 15;
        ia = r * 64 + cc * 4 + a;   ib = r * 64 + cc * 4 + b;
      }
      const float x = S[ia * 64 + cl];
      const float y = S[ib * 64 + cl];
      S[ia * 64 + cl] =  cth * x + sth * y;   // U[ia,ia]=c, U[ia,ib]=s
      S[ib * 64 + cl] = -sth * x + cth * y;   // U[ib,ia]=-s, U[ib,ib]=c
    }
    __syncthreads();
  }

  for (int idx = tid; idx < 1024 * 64; idx += 256) {
    const int row = idx >> 6, cl = idx & 63;
    W[row * DDIM + c0 + cl] = S[idx];
  }
}

// ---------------------------------------------------------------------------
// Kernel 2/3: C = A * B (TRANSB=0) or C = A * B^T (TRANSB=1), all 1024^3 f32.
// 128x128 block tile, 256 threads = 8 wave32 waves; wave w owns rows
// [16w,16w+16) x 128 cols -> 8 accumulators; fp32 WMMA 16x16x4.
// ---------------------------------------------------------------------------
template <bool TRANSB>
__global__ __launch_bounds__(256) void wmma_gemm_1024(
    const float* __restrict__ A, const float* __restrict__ B,
    float* __restrict__ C) {
  __shared__ float As[128][18];   // 128 rows x 16 k (stride 18: conflict-free)
  __shared__ float Bs[16][128];   // logical [k][n]

  const int tid  = threadIdx.x;
  const int lane = tid & 31;
  const int wv   = tid >> 5;      // wave 0..7
  const int half = lane >> 4;     // 0: lanes 0-15, 1: lanes 16-31
  const int l16  = lane & 15;
  const int m0 = blockIdx.y * 128;
  const int n0 = blockIdx.x * 128;

  v8f acc[8];
#pragma unroll
  for (int i = 0; i < 8; ++i)
#pragma unroll
    for (int e = 0; e < 8; ++e) acc[i][e] = 0.0f;

  for (int k0 = 0; k0 < DDIM; k0 += 16) {
    // stage A tile: 128 x 16
#pragma unroll
    for (int q = tid; q < 512; q += 256) {
      const int row = q >> 2, c4 = (q & 3) << 2;
      const float4 v = *(const float4*)(A + (size_t)(m0 + row) * DDIM + k0 + c4);
      As[row][c4 + 0] = v.x; As[row][c4 + 1] = v.y;
      As[row][c4 + 2] = v.z; As[row][c4 + 3] = v.w;
    }
    // stage B tile: logical [k][n] = 16 x 128
    if (!TRANSB) {
#pragma unroll
      for (int q = tid; q < 512; q += 256) {
        const int kk = q >> 5, c4 = (q & 31) << 2;
        const float4 v = *(const float4*)(B + (size_t)(k0 + kk) * DDIM + n0 + c4);
        Bs[kk][c4 + 0] = v.x; Bs[kk][c4 + 1] = v.y;
        Bs[kk][c4 + 2] = v.z; Bs[kk][c4 + 3] = v.w;
      }
    } else {  // B_logical[k][n] = B[n][k] : load rows of B, scatter-transpose
#pragma unroll
      for (int q = tid; q < 512; q += 256) {
        const int n = q >> 2, c4 = (q & 3) << 2;
        const float4 v = *(const float4*)(B + (size_t)(n0 + n) * DDIM + k0 + c4);
        Bs[c4 + 0][n] = v.x; Bs[c4 + 1][n] = v.y;
        Bs[c4 + 2][n] = v.z; Bs[c4 + 3][n] = v.w;
      }
    }
    __syncthreads();

#pragma unroll
    for (int kk = 0; kk < 16; kk += 4) {
      const int koff = kk + 2 * half;  // A/B 16x4 f32 layout: lanes 0-15 K=0,1;
      v2f a;                           //                      lanes 16-31 K=2,3
      a.x = As[wv * 16 + l16][koff];
      a.y = As[wv * 16 + l16][koff + 1];
#pragma unroll
      for (int nt = 0; nt < 8; ++nt) {
        v2f b;
        b.x = Bs[koff][nt * 16 + l16];
        b.y = Bs[koff + 1][nt * 16 + l16];
        acc[nt] = __builtin_amdgcn_wmma_f32_16x16x4_f32(
            false, a, false, b, (short)0, acc[nt], false, false);
      }
    }
    __syncthreads();
  }

  // D layout: VGPR r holds M = r + 8*half, N = lane&15
#pragma unroll
  for (int nt = 0; nt < 8; ++nt)
#pragma unroll
    for (int r = 0; r < 8; ++r) {
      const int row = m0 + wv * 16 + 8 * half + r;
      const int col = n0 + nt * 16 + l16;
      C[(size_t)row * DDIM + col] = acc[nt][r];
    }
}

extern "C" void kernel_launch(void* const* d_in, const int* in_sizes, int n_in,
                              void* d_out, int out_size, void* d_ws,
                              size_t ws_size, hipStream_t stream) {
  (void)in_sizes; (void)n_in; (void)out_size; (void)ws_size;
  const float* rho    = (const float*)d_in[0];  // input_state [1024,1024] f32
  const float* thetas = (const float*)d_in[1];  // [18] f32
  // d_in[2..4] (M0s/M1s/M2s) are deterministic masks, reproduced analytically.

  float* W = (float*)d_ws;                       // 4 MB
  float* T = (float*)d_ws + (size_t)DDIM * DDIM; // 4 MB
  float* out = (float*)d_out;

  // 1) W = U17 * ... * U0  (16 column stripes, 256KB LDS each)
  build_w_kernel<<<16, 256, 1024 * 64 * sizeof(float), stream>>>(thetas, W);

  // 2) T = W * rho ; 3) out = T * W^T   (fp32 WMMA GEMMs)
  dim3 grid(DDIM / 128, DDIM / 128);
  wmma_gemm_1024<false><<<grid, 256, 0, stream>>>(W, rho, T);
  wmma_gemm_1024<true ><<<grid, 256, 0, stream>>>(T, W, out);
}